// SimpleRefiner_24541443129997
// MI455X (gfx1250) — compile-verified
//
#include <hip/hip_runtime.h>
#include <hip/hip_bf16.h>

typedef __attribute__((ext_vector_type(2))) float v2f;
typedef __attribute__((ext_vector_type(8))) float v8f;

#define FDIM 128

// ---------------------------------------------------------------------------
// Stage 1: edge scatter. One wave32 per edge; lane L handles features
// [4L, 4L+4) -> float4 gather from x[src], 4 native fp32 atomics into
// nei_sum[dst]; lane 0 counts degree. x (51MB) + nei_sum (51MB) are
// L2-resident (192MB L2), so this runs at L2 atomic throughput.
// ---------------------------------------------------------------------------
__global__ __launch_bounds__(256) void sage_scatter(
    const float* __restrict__ x,
    const int* __restrict__ src,
    const int* __restrict__ dst,
    float* __restrict__ nei_sum,
    float* __restrict__ deg,
    int E)
{
    int wave = blockIdx.x * (blockDim.x >> 5) + (threadIdx.x >> 5);
    int lane = threadIdx.x & 31;
    if (wave >= E) return;

    int s = src[wave];
    int d = dst[wave];

    const float4* xr = (const float4*)(x + (size_t)s * FDIM);
    float4 v = xr[lane];

    float* o = nei_sum + (size_t)d * FDIM + lane * 4;
    unsafeAtomicAdd(o + 0, v.x);
    unsafeAtomicAdd(o + 1, v.y);
    unsafeAtomicAdd(o + 2, v.z);
    unsafeAtomicAdd(o + 3, v.w);

    if (lane == 0) unsafeAtomicAdd(deg + d, 1.0f);
}

// ---------------------------------------------------------------------------
// Stage 2: fused dual-GEMM + bias + mask + ReLU using V_WMMA_F32_16X16X4_F32.
// One wave per 16-node row tile. For each 16-wide output column tile:
//   acc = sum_k x[m,k]*Ws[n,k] + sum_k (scale[m]*nei_sum[m,k])*Wn[n,k]
// Fragment layout (f32, K=4): lane -> idx=lane%16, half=lane/16,
//   elem j of float2 = M[idx][kbase + 2*half + j]  (both A and B).
// ---------------------------------------------------------------------------
__global__ __launch_bounds__(256) void sage_gemm(
    const float* __restrict__ x,
    const float* __restrict__ nei_sum,
    const float* __restrict__ deg,
    const float* __restrict__ Ws,
    const float* __restrict__ bs,
    const float* __restrict__ Wn,
    const float* __restrict__ bn,
    float* __restrict__ out,
    int ntiles)
{
    int wave = blockIdx.x * (blockDim.x >> 5) + (threadIdx.x >> 5);
    if (wave >= ntiles) return;           // uniform per wave: EXEC stays all-1s
    int lane = threadIdx.x & 31;
    int half = lane >> 4;
    int idx  = lane & 15;
    int m0   = wave * 16;

    const float* xr = x       + (size_t)(m0 + idx) * FDIM;
    const float* nr = nei_sum + (size_t)(m0 + idx) * FDIM;

    float dg    = deg[m0 + idx];
    float scale = dg > 0.0f ? 1.0f / dg : 0.0f;   // masks isolated nodes

    // has-neighbor flags for the 8 output rows this lane's C fragment covers
    float has[8];
#pragma unroll
    for (int r = 0; r < 8; ++r)
        has[r] = (deg[m0 + r + 8 * half] > 0.0f) ? 1.0f : 0.0f;

    for (int ct = 0; ct < 8; ++ct) {
        const float* wsr = Ws + (size_t)(ct * 16 + idx) * FDIM;
        const float* wnr = Wn + (size_t)(ct * 16 + idx) * FDIM;

        v8f acc = {};

        // self path: x @ Ws^T
#pragma unroll
        for (int k = 0; k < FDIM; k += 4) {
            int kk = k + 2 * half;
            v2f a = *(const v2f*)(xr + kk);
            v2f b = *(const v2f*)(wsr + kk);
            acc = __builtin_amdgcn_wmma_f32_16x16x4_f32(
                false, a, false, b, (short)0, acc, false, false);
        }
        // neighbor path: (scale * nei_sum) @ Wn^T
#pragma unroll
        for (int k = 0; k < FDIM; k += 4) {
            int kk = k + 2 * half;
            v2f a = *(const v2f*)(nr + kk);
            a.x *= scale;
            a.y *= scale;
            v2f b = *(const v2f*)(wnr + kk);
            acc = __builtin_amdgcn_wmma_f32_16x16x4_f32(
                false, a, false, b, (short)0, acc, false, false);
        }

        // bias + masked neighbor bias + ReLU, then store.
        // C/D layout: VGPR r, lane -> M = r + 8*half, N = idx.
#pragma unroll
        for (int r = 0; r < 8; ++r) {
            int m = m0 + r + 8 * half;
            int n = ct * 16 + idx;
            float v = acc[r] + bs[n] + has[r] * bn[n];
            out[(size_t)m * FDIM + n] = v > 0.0f ? v : 0.0f;
        }
    }
}

// ---------------------------------------------------------------------------
extern "C" void kernel_launch(void* const* d_in, const int* in_sizes, int n_in,
                              void* d_out, int out_size, void* d_ws, size_t ws_size,
                              hipStream_t stream) {
    const float* x  = (const float*)d_in[0];
    const int*   src = (const int*)d_in[1];
    const int*   dst = (const int*)d_in[2];
    const float* Ws = (const float*)d_in[3];
    const float* bs = (const float*)d_in[4];
    const float* Wn = (const float*)d_in[5];
    const float* bn = (const float*)d_in[6];
    float* out = (float*)d_out;

    int N = in_sizes[0] / FDIM;
    int E = in_sizes[1];

    float* nei_sum = (float*)d_ws;
    float* deg     = nei_sum + (size_t)N * FDIM;

    // zero the accumulators (workspace is poisoned, must re-init every call)
    hipMemsetAsync(d_ws, 0, ((size_t)N * FDIM + N) * sizeof(float), stream);

    // scatter: one wave per edge, 8 waves per 256-thread block
    int scatter_blocks = (E + 7) / 8;
    sage_scatter<<<scatter_blocks, 256, 0, stream>>>(x, src, dst, nei_sum, deg, E);

    // fused GEMM: one wave per 16-row tile
    int ntiles = (N + 15) / 16;   // 6250 for N=100000 (exact)
    int gemm_blocks = (ntiles + 7) / 8;
    sage_gemm<<<gemm_blocks, 256, 0, stream>>>(x, nei_sum, deg, Ws, bs, Wn, bn, out, ntiles);
}